// encoder_36575941492983
// MI455X (gfx1250) — compile-verified
//
#include <hip/hip_runtime.h>
#include <cmath>

typedef __attribute__((ext_vector_type(2))) float v2f;
typedef __attribute__((ext_vector_type(4))) float v4f;
typedef __attribute__((ext_vector_type(8))) float v8f;
typedef __attribute__((ext_vector_type(4))) unsigned int u32x4;
typedef __attribute__((ext_vector_type(4))) int i32x4;
typedef __attribute__((ext_vector_type(8))) int i32x8;

#define Bsz  1024
#define Tm1  127
#define Ndim 256
#define Hdim 256

// LDS tile row stride (floats): 16 data + 4 pad -> bank-conflict-free fragment
// reads AND 16B-aligned rows so TDM pad (interval 16 DW, amount 4 DW) matches.
#define LDSTR 20
#define BUFSZ (128 * LDSTR)   // one LDS buffer: 128 rows x 20 floats

// ---------------------------------------------------------------------------
// TDM: issue a 2D tile load (tile_k=16 cols x `rows` rows of f32) into LDS.
// Descriptor per CDNA5 ISA ch.8: group0 = {ctrl, lds_addr, gaddr_lo, gaddr_hi|type},
// group1 = {flags/pad, dims, tile dims, strides}. pad: after every 16 DWORDs
// insert 4 DWORDs -> LDS row stride 20 floats. Tensor dims set huge (no OOB).
// ---------------------------------------------------------------------------
__device__ __forceinline__ void tdm_load_2d(unsigned int lds_byte,
                                            const float* gptr,
                                            unsigned int rows,
                                            unsigned long long row_stride_elems)
{
    unsigned long long ga = (unsigned long long)(size_t)gptr;
    u32x4 g0;
    g0[0] = 1u;                                   // count=1, user mode
    g0[1] = lds_byte;                             // lds_addr (bytes)
    g0[2] = (unsigned int)ga;                     // global_addr[31:0]
    g0[3] = (unsigned int)(ga >> 32) | (2u << 30);// global_addr[56:32] | type=2
    i32x8 g1;
    g1[0] = (int)((2u << 16)      // data_size = 4B
                | (1u << 20)      // pad_enable
                | (3u << 22)      // pad_interval: 16 DWORDs
                | (3u << 25));    // pad_amount:   4 DWORDs
    g1[1] = 0;                                    // atomic_barrier=0, dim0 lo16=0
    g1[2] = (int)0x4000u;                         // tensor_dim0 = 0x4000_0000
    g1[3] = (int)((16u << 16) | 0x4000u);         // tile_dim0=16 | tensor_dim1 hi
    g1[4] = (int)rows;                            // tile_dim1 = rows (tile_dim2=0)
    g1[5] = (int)(unsigned int)row_stride_elems;  // tensor_dim0_stride lo32
    g1[6] = (int)(unsigned int)(row_stride_elems >> 32); // stride hi16 (dim1_stride=0)
    g1[7] = 0;
    i32x4 z4 = {};
    i32x8 z8 = {};
    // 6-arg toolchain variant: (g0, g1, g2, g3, extra, cpol)
    __builtin_amdgcn_tensor_load_to_lds(g0, g1, z4, z4, z8, 0);
}

// ---------------------------------------------------------------------------
// One K=256 GEMM phase: acc[g][mt] += A(128 x 256) x W(j,k)^T tile, TDM-fed,
// LDS double-buffered, fp32 WMMA 16x16x4.
//   A rows   = batch rows b0..b0+127, row stride ldA
//   W rows   = gate rows g*256 + jh0 + jj (4 groups of 32), row stride 256
// Wave layout: 8 waves = 4 (M, 32 rows each) x 2 (jh, 16 cols each).
// ---------------------------------------------------------------------------
__device__ __forceinline__ void gemm_phase(
    const float* __restrict__ Abase, size_t ldA,
    const float* __restrict__ Wmat,
    int b0, int jh0, int wv, int lane, int wm, int wj,
    v8f acc[4][2], float* A_sh, float* W_sh)
{
    const int l16  = lane & 15;
    const int half = lane >> 4;

    auto issue = [&](int s) {
        const int kb  = s * 16;
        const int buf = (s & 1) * BUFSZ;
        tdm_load_2d((unsigned int)(size_t)&A_sh[buf],
                    Abase + (size_t)b0 * ldA + kb, 128u, (unsigned long long)ldA);
        #pragma unroll
        for (int g = 0; g < 4; ++g)
            tdm_load_2d((unsigned int)(size_t)&W_sh[buf + g * 32 * LDSTR],
                        Wmat + (size_t)(g * 256 + jh0) * 256 + kb, 32u, 256ull);
    };

    if (wv == 0) issue(0);

    #pragma unroll 1
    for (int s = 0; s < 16; ++s) {
        if (wv == 0) {
            if (s + 1 < 16) { issue(s + 1); __builtin_amdgcn_s_wait_tensorcnt(5); }
            else            {               __builtin_amdgcn_s_wait_tensorcnt(0); }
        }
        __syncthreads();
        const float* As = &A_sh[(s & 1) * BUFSZ];
        const float* Ws = &W_sh[(s & 1) * BUFSZ];
        #pragma unroll
        for (int kk = 0; kk < 16; kk += 4) {
            v2f afr[2];
            #pragma unroll
            for (int mt = 0; mt < 2; ++mt) {
                const int row = wm * 32 + mt * 16 + l16;
                afr[mt] = *(const v2f*)&As[row * LDSTR + kk + 2 * half];
            }
            #pragma unroll
            for (int g = 0; g < 4; ++g) {
                const int jcol = g * 32 + wj * 16 + l16;
                const float* wp = &Ws[jcol * LDSTR + kk];
                v2f bfr;
                bfr.x = wp[half];        // V0: K = kk   | kk+1
                bfr.y = wp[2 + half];    // V1: K = kk+2 | kk+3
                #pragma unroll
                for (int mt = 0; mt < 2; ++mt)
                    acc[g][mt] = __builtin_amdgcn_wmma_f32_16x16x4_f32(
                        false, afr[mt], false, bfr, (short)0, acc[g][mt],
                        false, false);
            }
        }
        __syncthreads();
    }
}

// ---------------------------------------------------------------------------
// Kernel 0: zero LSTM state buffers
// ---------------------------------------------------------------------------
__global__ void zero_ws_kernel(float* __restrict__ p, int n) {
    int i = blockIdx.x * blockDim.x + threadIdx.x;
    if (i < n) p[i] = 0.0f;
}

// ---------------------------------------------------------------------------
// Kernel 1: time-invariant attention + weighted output.
// (h@wh + c@wc) and b_attn are constant along the softmax axis -> cancel, so
// attn[b,n] = softmax_n( sum_t x[b,t,n]*wx[t] ), computed once.
// ---------------------------------------------------------------------------
__global__ __launch_bounds__(256) void attn_kernel(
    const float* __restrict__ x, const float* __restrict__ Wattn,
    float* __restrict__ wout)
{
    __shared__ float wx_s[128];
    __shared__ float red[256];
    const int b = blockIdx.x, tid = threadIdx.x;

    if (tid < Tm1) wx_s[tid] = Wattn[2 * Hdim + tid];
    __syncthreads();

    const float* xb = x + (size_t)b * Tm1 * Ndim + tid;
    float s = 0.0f;
    #pragma unroll 4
    for (int t = 0; t < Tm1; ++t) s += xb[(size_t)t * Ndim] * wx_s[t];

    red[tid] = s; __syncthreads();
    for (int off = 128; off > 0; off >>= 1) {
        if (tid < off) red[tid] = fmaxf(red[tid], red[tid + off]);
        __syncthreads();
    }
    float mx = red[0]; __syncthreads();
    float e = __expf(s - mx);
    red[tid] = e; __syncthreads();
    for (int off = 128; off > 0; off >>= 1) {
        if (tid < off) red[tid] += red[tid + off];
        __syncthreads();
    }
    const float a = e / red[0];

    float* wo = wout + (size_t)b * Tm1 * Ndim + tid;
    #pragma unroll 4
    for (int t = 0; t < Tm1; ++t) wo[(size_t)t * Ndim] = a * xb[(size_t)t * Ndim];
}

// ---------------------------------------------------------------------------
// Kernel 2 (XG path, fully parallel over t): XG[t,b,j] = weighted_t @ W_ih^T
// grid (8, 8, 127)
// ---------------------------------------------------------------------------
__global__ __launch_bounds__(256) void xg_gemm_kernel(
    const float* __restrict__ weighted, const float* __restrict__ Wih,
    float* __restrict__ XG)
{
    __shared__ float A_sh[2 * BUFSZ];
    __shared__ float W_sh[2 * BUFSZ];
    const int tid = threadIdx.x, lane = tid & 31, wv = tid >> 5;
    const int wm = wv & 3, wj = wv >> 2;
    const int half = lane >> 4, l16 = lane & 15;
    const int b0 = blockIdx.x * 128, jh0 = blockIdx.y * 32, t = blockIdx.z;

    v8f acc[4][2];
    const v8f vz = {};
    #pragma unroll
    for (int g = 0; g < 4; ++g)
        #pragma unroll
        for (int mt = 0; mt < 2; ++mt) acc[g][mt] = vz;

    gemm_phase(weighted + (size_t)t * Ndim, (size_t)Tm1 * Ndim, Wih,
               b0, jh0, wv, lane, wm, wj, acc, A_sh, W_sh);

    const int jh = jh0 + wj * 16 + l16;
    #pragma unroll
    for (int mt = 0; mt < 2; ++mt)
        #pragma unroll
        for (int v = 0; v < 8; ++v) {
            const int b = b0 + wm * 32 + mt * 16 + v + 8 * half;
            float* dst = XG + ((size_t)t * Bsz + b) * 1024 + jh;
            #pragma unroll
            for (int g = 0; g < 4; ++g) dst[g * 256] = acc[g][mt][v];
        }
}

// ---------------------------------------------------------------------------
// Kernel 3a (XG path): recurrent step, K=256 only (h @ W_hh^T) + activations
// ---------------------------------------------------------------------------
__global__ __launch_bounds__(256) void lstm_step_xg_kernel(
    int t, const float* __restrict__ XG,
    const float* __restrict__ Whh,
    const float* __restrict__ bih, const float* __restrict__ bhh,
    const float* __restrict__ h_in, float* __restrict__ h_out,
    float* __restrict__ c, float* __restrict__ enc)
{
    __shared__ float A_sh[2 * BUFSZ];
    __shared__ float W_sh[2 * BUFSZ];
    const int tid = threadIdx.x, lane = tid & 31, wv = tid >> 5;
    const int wm = wv & 3, wj = wv >> 2;
    const int half = lane >> 4, l16 = lane & 15;
    const int b0 = blockIdx.x * 128, jh0 = blockIdx.y * 32;

    v8f acc[4][2];
    const v8f vz = {};
    #pragma unroll
    for (int g = 0; g < 4; ++g)
        #pragma unroll
        for (int mt = 0; mt < 2; ++mt) acc[g][mt] = vz;

    gemm_phase(h_in, (size_t)Hdim, Whh, b0, jh0, wv, lane, wm, wj, acc, A_sh, W_sh);

    const int jh = jh0 + wj * 16 + l16;
    float bias[4];
    #pragma unroll
    for (int g = 0; g < 4; ++g) bias[g] = bih[g * 256 + jh] + bhh[g * 256 + jh];

    #pragma unroll
    for (int mt = 0; mt < 2; ++mt)
        #pragma unroll
        for (int v = 0; v < 8; ++v) {
            const int b = b0 + wm * 32 + mt * 16 + v + 8 * half;
            const float* xg = XG + ((size_t)t * Bsz + b) * 1024 + jh;
            const float gi = acc[0][mt][v] + xg[0]       + bias[0];
            const float gf = acc[1][mt][v] + xg[256]     + bias[1];
            const float gg = acc[2][mt][v] + xg[512]     + bias[2];
            const float go = acc[3][mt][v] + xg[768]     + bias[3];
            const float si = 1.0f / (1.0f + __expf(-gi));
            const float sf = 1.0f / (1.0f + __expf(-gf));
            const float so = 1.0f / (1.0f + __expf(-go));
            const float tg = tanhf(gg);
            const size_t cidx = (size_t)b * Hdim + jh;
            const float cn = sf * c[cidx] + si * tg;
            const float hn = so * tanhf(cn);
            c[cidx] = cn;
            h_out[cidx] = hn;
            enc[(size_t)b * (Tm1 * Hdim) + (size_t)t * Hdim + jh] = hn;
        }
}

// ---------------------------------------------------------------------------
// Kernel 3b (fallback, small ws): fused step, K=512 (x-part + h-part)
// ---------------------------------------------------------------------------
__global__ __launch_bounds__(256) void lstm_step_fused_kernel(
    int t, const float* __restrict__ weighted,
    const float* __restrict__ Wih, const float* __restrict__ Whh,
    const float* __restrict__ bih, const float* __restrict__ bhh,
    const float* __restrict__ h_in, float* __restrict__ h_out,
    float* __restrict__ c, float* __restrict__ enc)
{
    __shared__ float A_sh[2 * BUFSZ];
    __shared__ float W_sh[2 * BUFSZ];
    const int tid = threadIdx.x, lane = tid & 31, wv = tid >> 5;
    const int wm = wv & 3, wj = wv >> 2;
    const int half = lane >> 4, l16 = lane & 15;
    const int b0 = blockIdx.x * 128, jh0 = blockIdx.y * 32;

    v8f acc[4][2];
    const v8f vz = {};
    #pragma unroll
    for (int g = 0; g < 4; ++g)
        #pragma unroll
        for (int mt = 0; mt < 2; ++mt) acc[g][mt] = vz;

    gemm_phase(weighted + (size_t)t * Ndim, (size_t)Tm1 * Ndim, Wih,
               b0, jh0, wv, lane, wm, wj, acc, A_sh, W_sh);
    gemm_phase(h_in, (size_t)Hdim, Whh,
               b0, jh0, wv, lane, wm, wj, acc, A_sh, W_sh);

    const int jh = jh0 + wj * 16 + l16;
    float bias[4];
    #pragma unroll
    for (int g = 0; g < 4; ++g) bias[g] = bih[g * 256 + jh] + bhh[g * 256 + jh];

    #pragma unroll
    for (int mt = 0; mt < 2; ++mt)
        #pragma unroll
        for (int v = 0; v < 8; ++v) {
            const int b = b0 + wm * 32 + mt * 16 + v + 8 * half;
            const float gi = acc[0][mt][v] + bias[0];
            const float gf = acc[1][mt][v] + bias[1];
            const float gg = acc[2][mt][v] + bias[2];
            const float go = acc[3][mt][v] + bias[3];
            const float si = 1.0f / (1.0f + __expf(-gi));
            const float sf = 1.0f / (1.0f + __expf(-gf));
            const float so = 1.0f / (1.0f + __expf(-go));
            const float tg = tanhf(gg);
            const size_t cidx = (size_t)b * Hdim + jh;
            const float cn = sf * c[cidx] + si * tg;
            const float hn = so * tanhf(cn);
            c[cidx] = cn;
            h_out[cidx] = hn;
            enc[(size_t)b * (Tm1 * Hdim) + (size_t)t * Hdim + jh] = hn;
        }
}

// ---------------------------------------------------------------------------
extern "C" void kernel_launch(void* const* d_in, const int* in_sizes, int n_in,
                              void* d_out, int out_size, void* d_ws, size_t ws_size,
                              hipStream_t stream) {
    (void)in_sizes; (void)n_in; (void)out_size;
    const float* x     = (const float*)d_in[0];
    const float* Wih   = (const float*)d_in[1];
    const float* Whh   = (const float*)d_in[2];
    const float* bih   = (const float*)d_in[3];
    const float* bhh   = (const float*)d_in[4];
    const float* Wattn = (const float*)d_in[5];
    // d_in[6] (b_attn) cancels inside softmax -> unused.

    float* out          = (float*)d_out;
    float* weighted_out = out;                             // (B, Tm1, N)
    float* enc_out      = out + (size_t)Bsz * Tm1 * Ndim;  // (B, Tm1, H)

    float* wsf = (float*)d_ws;
    const size_t stateElems = (size_t)Bsz * Hdim;          // 262144
    float* cbuf = wsf;
    float* h0   = wsf + stateElems;
    float* h1   = wsf + 2 * stateElems;
    float* XG   = wsf + 3 * stateElems;                    // (Tm1, B, 4H)

    const size_t xgElems = (size_t)Tm1 * Bsz * 4 * Hdim;
    const bool use_xg = ws_size >= (3 * stateElems + xgElems) * sizeof(float);

    const int zn = 3 * (int)stateElems;
    zero_ws_kernel<<<(zn + 255) / 256, 256, 0, stream>>>(wsf, zn);

    attn_kernel<<<Bsz, 256, 0, stream>>>(x, Wattn, weighted_out);

    if (use_xg) {
        // time-parallel half of the GEMM work: saturates the chip in one launch
        xg_gemm_kernel<<<dim3(8, 8, Tm1), 256, 0, stream>>>(weighted_out, Wih, XG);
        for (int t = 0; t < Tm1; ++t) {
            const float* hin = (t & 1) ? h1 : h0;
            float*       hout= (t & 1) ? h0 : h1;
            lstm_step_xg_kernel<<<dim3(8, 8), 256, 0, stream>>>(
                t, XG, Whh, bih, bhh, hin, hout, cbuf, enc_out);
        }
    } else {
        for (int t = 0; t < Tm1; ++t) {
            const float* hin = (t & 1) ? h1 : h0;
            float*       hout= (t & 1) ? h0 : h1;
            lstm_step_fused_kernel<<<dim3(8, 8), 256, 0, stream>>>(
                t, weighted_out, Wih, Whh, bih, bhh, hin, hout, cbuf, enc_out);
        }
    }
}